// CrossEntropyLossWithGaussianSmoothedLabels_6262062318150
// MI455X (gfx1250) — compile-verified
//
#include <hip/hip_runtime.h>
#include <stdint.h>

#define NUM_CLASSES   722
#define ROWS_TOTAL    (32 * 2048)          // 65536 rows
#define WAVES_PER_BLK 8
#define ROWS_PER_WAVE 8
#define NBLOCKS       (ROWS_TOTAL / (WAVES_PER_BLK * ROWS_PER_WAVE))  // 1024
#define F2_PER_ROW    361                  // 722 floats = 361 float2
#define ASYNC_PER_ROW 12                   // ceil(361 / 32) b64 wave-loads per row

#define L2E  1.44269504088896340736f       // log2(e)
#define LN2  0.69314718055994530942f
// final smoothed-label weights by |c - target| (target itself overwritten to 1.0)
#define W1   0.60653065971263342360f       // exp(-2/4)
#define W2   0.36787944117144232160f       // exp(-4/4)
#define W3   0.13533528323661269189f       // exp(-8/4)

// Low 32 bits of a flat pointer into LDS == byte offset within the
// workgroup's LDS allocation (aperture lives in addr[63:32]).
__device__ __forceinline__ unsigned lds_off32(const void* p) {
    return (unsigned)(unsigned long long)(uintptr_t)p;
}

// Issue one row (2888 B) as 12 wave-wide async b64 copies global -> LDS.
// Tail lanes clamp to the last float2 (duplicate write of identical data).
__device__ __forceinline__ void async_load_row(unsigned lds_base,
                                               const float* __restrict__ grow,
                                               int lane) {
#pragma unroll
    for (int k = 0; k < ASYNC_PER_ROW; ++k) {
        int e = lane + 32 * k;
        if (e > F2_PER_ROW - 1) e = F2_PER_ROW - 1;
        unsigned long long ga = (unsigned long long)(uintptr_t)grow
                              + (unsigned long long)e * 8ull;
        unsigned la = lds_base + (unsigned)e * 8u;
        asm volatile("global_load_async_to_lds_b64 %0, %1, off"
                     :: "v"(la), "v"(ga)
                     : "memory");
    }
}

__global__ __launch_bounds__(256) void
ce_gauss_smooth_kernel(const float* __restrict__ pred,
                       const int*   __restrict__ target,
                       float*       __restrict__ partial) {
    __shared__ float sh[WAVES_PER_BLK][2][NUM_CLASSES];
    __shared__ float wsum[WAVES_PER_BLK];

    const int tid  = threadIdx.x;
    const int lane = tid & 31;
    const int wave = tid >> 5;
    const int row0 = (blockIdx.x * WAVES_PER_BLK + wave) * ROWS_PER_WAVE;

    const unsigned lb0 = lds_off32(&sh[wave][0][0]);
    const unsigned lb1 = lds_off32(&sh[wave][1][0]);

    // Prime the pipeline: row0 -> buffer 0.
    async_load_row(lb0, pred + (size_t)row0 * NUM_CLASSES, lane);

    float acc = 0.0f;

    for (int i = 0; i < ROWS_PER_WAVE; ++i) {
        // Prefetch next row into the other buffer, then wait for current row.
        if (i + 1 < ROWS_PER_WAVE) {
            async_load_row((i & 1) ? lb0 : lb1,
                           pred + (size_t)(row0 + i + 1) * NUM_CLASSES, lane);
            __builtin_amdgcn_s_wait_asynccnt(ASYNC_PER_ROW);
        } else {
            __builtin_amdgcn_s_wait_asynccnt(0);
        }
        asm volatile("" ::: "memory");   // keep LDS reads after the wait

        const float* rowbuf = &sh[wave][i & 1][0];
        const float2* rb2   = (const float2*)rowbuf;

        // Pass 1: row max (lane-local over ~23 elems, then wave32 xor-reduce).
        float m = -3.402823466e+38f;
#pragma unroll
        for (int k = 0; k < ASYNC_PER_ROW; ++k) {
            int e = lane + 32 * k;
            if (e < F2_PER_ROW) {
                float2 v = rb2[e];
                m = fmaxf(m, fmaxf(v.x, v.y));
            }
        }
#pragma unroll
        for (int off = 16; off; off >>= 1)
            m = fmaxf(m, __shfl_xor(m, off, 32));

        // Pass 2: sum of exp2((x - m) * log2e).
        float s = 0.0f;
#pragma unroll
        for (int k = 0; k < ASYNC_PER_ROW; ++k) {
            int e = lane + 32 * k;
            if (e < F2_PER_ROW) {
                float2 v = rb2[e];
                s += __builtin_amdgcn_exp2f((v.x - m) * L2E)
                   + __builtin_amdgcn_exp2f((v.y - m) * L2E);
            }
        }
#pragma unroll
        for (int off = 16; off; off >>= 1)
            s += __shfl_xor(s, off, 32);

        if (lane == 0) {
            const int tgt = target[row0 + i];
            const float lse = m + LN2 * __builtin_amdgcn_logf(s); // m + ln(sum)
            float W = 0.0f, dot = 0.0f;
#pragma unroll
            for (int d = -3; d <= 3; ++d) {
                int c = tgt + d;
                if ((unsigned)c < (unsigned)NUM_CLASSES) {
                    float wt = (d == 0) ? 1.0f
                             : (d == 1 || d == -1) ? W1
                             : (d == 2 || d == -2) ? W2 : W3;
                    W   += wt;
                    dot += wt * rowbuf[c];
                }
            }
            acc += W * lse - dot;
        }
    }

    if (lane == 0) wsum[wave] = acc;
    __syncthreads();
    if (tid == 0) {
        float t = 0.0f;
#pragma unroll
        for (int w = 0; w < WAVES_PER_BLK; ++w) t += wsum[w];
        partial[blockIdx.x] = t;
    }
}

__global__ __launch_bounds__(256) void
ce_reduce_kernel(const float* __restrict__ partial, float* __restrict__ out) {
    __shared__ float sm[256];
    const int tid = threadIdx.x;
    float t = 0.0f;
    for (int i = tid; i < NBLOCKS; i += 256) t += partial[i];
    sm[tid] = t;
    __syncthreads();
#pragma unroll
    for (int s = 128; s > 0; s >>= 1) {
        if (tid < s) sm[tid] += sm[tid + s];
        __syncthreads();
    }
    if (tid == 0) out[0] = sm[0] * (1.0f / (float)ROWS_TOTAL);
}

extern "C" void kernel_launch(void* const* d_in, const int* in_sizes, int n_in,
                              void* d_out, int out_size, void* d_ws, size_t ws_size,
                              hipStream_t stream) {
    const float* pred   = (const float*)d_in[0];   // (32, 2048, 722) f32
    const int*   target = (const int*)d_in[1];     // (32, 2048) i32
    float* out     = (float*)d_out;                // scalar f32
    float* partial = (float*)d_ws;                 // NBLOCKS floats of scratch

    ce_gauss_smooth_kernel<<<NBLOCKS, 256, 0, stream>>>(pred, target, partial);
    ce_reduce_kernel<<<1, 256, 0, stream>>>(partial, out);
}